// MambaBlock_1382979470260
// MI455X (gfx1250) — compile-verified
//
#include <hip/hip_runtime.h>
#include <math.h>

// ---------------------------------------------------------------------------
// Mamba block for MI455X (gfx1250, wave32, WMMA + async-to-LDS).
// GEMMs: V_WMMA_F32_16X16X4_F32, 16x64 strip per wave, software-pipelined
//        fragment loads (prefetch distance 1, immediate-offset addressing).
// Scan: double-buffered LDS chunks filled with global_load_async_to_lds_b128.
// ---------------------------------------------------------------------------

typedef __attribute__((ext_vector_type(2))) float v2f;
typedef __attribute__((ext_vector_type(8))) float v8f;

#define D_MODEL 512
#define D_STATE 16
#define D_CONV  4
#define D_INNER 1024
#define DT_RANK 32
#define B_SZ    4
#define L_SEQ   2048
#define BL      (B_SZ * L_SEQ)              // 8192 token rows
#define XDBL_W  (DT_RANK + 2 * D_STATE)     // 64

__device__ __forceinline__ float silu_f(float x) {
  return x / (1.0f + __expf(-x));
}
__device__ __forceinline__ float softplus_f(float x) {
  return (x > 20.0f) ? x : __logf(1.0f + __expf(x));
}

// CDNA5 async copy: global -> LDS, 16 bytes per lane, tracked by ASYNCcnt.
__device__ __forceinline__ void async_ld_b128(void* lds_dst, const void* gsrc) {
  asm volatile("global_load_async_to_lds_b128 %0, %1, off"
               :: "v"((unsigned int)(unsigned long long)lds_dst),
                  "v"((unsigned long long)gsrc)
               : "memory");
}
__device__ __forceinline__ void wait_async0() {
  asm volatile("s_wait_asynccnt 0x0" ::: "memory");
}

__device__ __forceinline__ v8f wmma_f32(v2f a, v2f b, v8f c) {
  return __builtin_amdgcn_wmma_f32_16x16x4_f32(false, a, false, b,
                                               (short)0, c, false, false);
}

// ---------------------------------------------------------------------------
// C[M,N] = A[M,K] (row-major, LDA) * W[N,K]^T (row-major, LDW)
// Block = 256 threads = 8 waves stacked in M; block tile 128 x 64.
// Each wave computes a 16x64 strip: A fragment loaded once per k-step and
// reused across 4 WMMAs into 4 independent accumulators. Fragment loads are
// software-pipelined one k-step ahead so WMMAs wait only on old loads.
// All strides are compile-time so B-strip addresses fold into IOFFSET.
//
// fp32 16x16x4 fragment layout (ISA 7.12.2):
//   A: lane r (r=lane&15), half=lane>>4: VGPR v holds A[r][k0 + 2*half + v]
//   B (= W^T): VGPR v holds W[ncol + r][k0 + 2*half + v]
//   C/D: acc[v] = D[m_base + v + 8*half][ncol + r]
// EPI==1: per-column bias + softplus (dt path).
// ---------------------------------------------------------------------------
template <int K, int LDA, int LDW, int LDC, int EPI>
__global__ __launch_bounds__(256) void wmma_gemm_nt(
    const float* __restrict__ A, const float* __restrict__ W,
    float* __restrict__ C, const float* __restrict__ bias) {
  const int tid  = threadIdx.x;
  const int lane = tid & 31;
  const int wave = tid >> 5;
  const int half = lane >> 4;
  const int r    = lane & 15;

  const int m_base = blockIdx.y * 128 + wave * 16;
  const int n0     = blockIdx.x * 64;

  const float* Ap = A + (size_t)(m_base + r) * LDA + 2 * half;
  const float* Wp = W + (size_t)(n0 + r) * LDW + 2 * half;

  // current-stage fragments (k0 = 0)
  v2f a_c  = *(const v2f*)(Ap);
  v2f b0_c = *(const v2f*)(Wp);
  v2f b1_c = *(const v2f*)(Wp + 16 * LDW);
  v2f b2_c = *(const v2f*)(Wp + 32 * LDW);
  v2f b3_c = *(const v2f*)(Wp + 48 * LDW);

  v8f acc0 = {0.f,0.f,0.f,0.f,0.f,0.f,0.f,0.f};
  v8f acc1 = acc0, acc2 = acc0, acc3 = acc0;

#pragma unroll 4
  for (int k0 = 0; k0 < K - 4; k0 += 4) {
    // prefetch next k-step before consuming current fragments
    v2f a_n  = *(const v2f*)(Ap + k0 + 4);
    v2f b0_n = *(const v2f*)(Wp + k0 + 4);
    v2f b1_n = *(const v2f*)(Wp + 16 * LDW + k0 + 4);
    v2f b2_n = *(const v2f*)(Wp + 32 * LDW + k0 + 4);
    v2f b3_n = *(const v2f*)(Wp + 48 * LDW + k0 + 4);

    acc0 = wmma_f32(a_c, b0_c, acc0);
    acc1 = wmma_f32(a_c, b1_c, acc1);
    acc2 = wmma_f32(a_c, b2_c, acc2);
    acc3 = wmma_f32(a_c, b3_c, acc3);

    a_c = a_n; b0_c = b0_n; b1_c = b1_n; b2_c = b2_n; b3_c = b3_n;
  }
  // final k-step
  acc0 = wmma_f32(a_c, b0_c, acc0);
  acc1 = wmma_f32(a_c, b1_c, acc1);
  acc2 = wmma_f32(a_c, b2_c, acc2);
  acc3 = wmma_f32(a_c, b3_c, acc3);

  v8f accs[4] = {acc0, acc1, acc2, acc3};
#pragma unroll
  for (int j = 0; j < 4; ++j) {
    const int   col  = n0 + 16 * j + r;
    const float bval = (EPI == 1) ? bias[col] : 0.0f;
#pragma unroll
    for (int v = 0; v < 8; ++v) {
      const int m   = m_base + v + 8 * half;
      float     val = accs[j][v];
      if (EPI == 1) val = softplus_f(val + bval);
      C[(size_t)m * LDC + col] = val;
    }
  }
}

// ---------------------------------------------------------------------------
// Depthwise causal conv1d (width 4) + bias + SiLU over the xc half of xz.
// ---------------------------------------------------------------------------
__global__ __launch_bounds__(256) void conv_silu_kernel(
    const float* __restrict__ xz, const float* __restrict__ conv_w,
    const float* __restrict__ conv_b, float* __restrict__ xc_out) {
  const int idx = blockIdx.x * 256 + threadIdx.x;  // over BL * D_INNER
  const int d   = idx & (D_INNER - 1);
  const int bl  = idx >> 10;               // global token row b*L + l
  const int l   = bl & (L_SEQ - 1);

  float s = conv_b[d];
#pragma unroll
  for (int j = 0; j < D_CONV; ++j) {
    const int lt = l - (D_CONV - 1) + j;
    if (lt >= 0)
      s += conv_w[d * D_CONV + j] *
           xz[(size_t)(bl + (lt - l)) * (2 * D_INNER) + d];
  }
  xc_out[idx] = silu_f(s);
}

// ---------------------------------------------------------------------------
// Selective scan, fused skip + gate epilogue:
//   h[n] = exp(dt*A[d,n])*h[n] + dt*x*B[n];  y = sum_n h[n]*C[n]
//   yg   = (y + x*D_skip[d]) * silu(z)
// 4 lanes per channel (4 states each) -> shfl_xor reduce within the quad.
// dt/xc/z/B/C staged into double-buffered LDS chunks of 32 steps via
// global_load_async_to_lds_b128, so global latency hides behind compute.
// Grid: (D_INNER/64, B_SZ); block 256 threads = 64 channels x 4 subs.
// ---------------------------------------------------------------------------
#define SC_CHUNK 32
__global__ __launch_bounds__(256) void scan_kernel(
    const float* __restrict__ xc, const float* __restrict__ dt,
    const float* __restrict__ xdbl, const float* __restrict__ xz,
    const float* __restrict__ A_log, const float* __restrict__ Dskip,
    float* __restrict__ yg) {
  __shared__ __align__(16) float sDT[2][SC_CHUNK][64];
  __shared__ __align__(16) float sXC[2][SC_CHUNK][64];
  __shared__ __align__(16) float sZ [2][SC_CHUNK][64];
  __shared__ __align__(16) float sBC[2][SC_CHUNK][32];  // [., 0:16)=B [16:32)=C

  const int    tid  = threadIdx.x;
  const int    sub  = tid & 3;         // which 4-state slice of D_STATE
  const int    dloc = tid >> 2;        // 0..63
  const int    d0   = blockIdx.x * 64;
  const int    d    = d0 + dloc;
  const size_t row0 = (size_t)blockIdx.y * L_SEQ;  // first token row of batch

  // Stage one chunk [c0, c0+32) into LDS buffer `buf` (async).
  auto stage = [&](int buf, int c0) {
#pragma unroll
    for (int j = 0; j < 2; ++j) {
      const int    g    = tid + j * 256;       // group of 4 floats
      const int    step = g >> 4;              // 0..31
      const int    dof  = (g & 15) * 4;        // 0..60
      const size_t row  = row0 + c0 + step;
      async_ld_b128(&sDT[buf][step][dof], dt + row * D_INNER + d0 + dof);
      async_ld_b128(&sXC[buf][step][dof], xc + row * D_INNER + d0 + dof);
      async_ld_b128(&sZ [buf][step][dof],
                    xz + row * (2 * D_INNER) + D_INNER + d0 + dof);
    }
    {
      const int g    = tid;                    // 256 groups of 4 floats
      const int step = g >> 3;                 // 0..31
      const int qof  = (g & 7) * 4;            // 0..28
      async_ld_b128(&sBC[buf][step][qof],
                    xdbl + (row0 + c0 + step) * XDBL_W + DT_RANK + qof);
    }
  };

  float a[4], h[4];
#pragma unroll
  for (int j = 0; j < 4; ++j) {
    a[j] = -__expf(A_log[d * D_STATE + sub * 4 + j]);
    h[j] = 0.0f;
  }
  const float dsk = Dskip[d];

  stage(0, 0);
  wait_async0();
  __syncthreads();

  int cur = 0;
  for (int c0 = 0; c0 < L_SEQ; c0 += SC_CHUNK) {
    if (c0 + SC_CHUNK < L_SEQ) stage(cur ^ 1, c0 + SC_CHUNK);

    for (int t = 0; t < SC_CHUNK; ++t) {
      const float dtt = sDT[cur][t][dloc];
      const float xt  = sXC[cur][t][dloc];
      const float dx  = dtt * xt;
      float       y   = 0.0f;
#pragma unroll
      for (int j = 0; j < 4; ++j) {
        const int   n  = sub * 4 + j;
        const float dA = __expf(dtt * a[j]);
        h[j] = dA * h[j] + dx * sBC[cur][t][n];
        y += h[j] * sBC[cur][t][16 + n];
      }
      // reduce across the 4 lanes of this channel's quad
      y += __shfl_xor(y, 1);
      y += __shfl_xor(y, 2);
      if (sub == 0) {
        const float zt = sZ[cur][t][dloc];
        yg[(row0 + c0 + t) * D_INNER + d] = (y + xt * dsk) * silu_f(zt);
      }
    }

    wait_async0();     // prefetched chunk fully in LDS
    __syncthreads();   // everyone done reading old buffer
    cur ^= 1;
  }
}

// ---------------------------------------------------------------------------
// Host-side orchestration. Workspace layout (floats):
//   xz   : BL * 2048   (67.1 MB)
//   xc   : BL * 1024   (33.6 MB)
//   xdbl : BL * 64     ( 2.1 MB)
//   dt   : BL * 1024   (33.6 MB)
//   yg   : BL * 1024   (33.6 MB)   total ~170 MB
// ---------------------------------------------------------------------------
extern "C" void kernel_launch(void* const* d_in, const int* in_sizes, int n_in,
                              void* d_out, int out_size, void* d_ws,
                              size_t ws_size, hipStream_t stream) {
  (void)in_sizes; (void)n_in; (void)out_size; (void)ws_size;
  const float* x      = (const float*)d_in[0];
  const float* W_in   = (const float*)d_in[1];
  const float* conv_w = (const float*)d_in[2];
  const float* conv_b = (const float*)d_in[3];
  const float* W_x    = (const float*)d_in[4];
  const float* W_dt   = (const float*)d_in[5];
  const float* b_dt   = (const float*)d_in[6];
  const float* A_log  = (const float*)d_in[7];
  const float* D_skip = (const float*)d_in[8];
  const float* W_out  = (const float*)d_in[9];
  float*       out    = (float*)d_out;

  float* ws   = (float*)d_ws;
  float* xz   = ws;
  float* xc   = xz   + (size_t)BL * (2 * D_INNER);
  float* xdbl = xc   + (size_t)BL * D_INNER;
  float* dt   = xdbl + (size_t)BL * XDBL_W;
  float* yg   = dt   + (size_t)BL * D_INNER;

  const dim3 blk(256);

  // 1) xz = x @ W_in^T : (8192x512)x(2048x512)^T
  wmma_gemm_nt<D_MODEL, D_MODEL, D_MODEL, 2 * D_INNER, 0>
      <<<dim3((2 * D_INNER) / 64, BL / 128), blk, 0, stream>>>(
          x, W_in, xz, nullptr);

  // 2) depthwise causal conv + SiLU on xc half
  conv_silu_kernel<<<(BL * D_INNER) / 256, blk, 0, stream>>>(xz, conv_w,
                                                             conv_b, xc);

  // 3) x_dbl = xc @ W_x^T : (8192x1024)x(64x1024)^T
  wmma_gemm_nt<D_INNER, D_INNER, D_INNER, XDBL_W, 0>
      <<<dim3(XDBL_W / 64, BL / 128), blk, 0, stream>>>(
          xc, W_x, xdbl, nullptr);

  // 4) dt = softplus(x_dbl[:, :32] @ W_dt^T + b_dt) : (8192x32)x(1024x32)^T
  wmma_gemm_nt<DT_RANK, XDBL_W, DT_RANK, D_INNER, 1>
      <<<dim3(D_INNER / 64, BL / 128), blk, 0, stream>>>(
          xdbl, W_dt, dt, b_dt);

  // 5) selective scan fused with skip + gate -> yg
  scan_kernel<<<dim3(D_INNER / 64, B_SZ), blk, 0, stream>>>(
      xc, dt, xdbl, xz, A_log, D_skip, yg);

  // 6) out = yg @ W_out^T : (8192x1024)x(512x1024)^T
  wmma_gemm_nt<D_INNER, D_INNER, D_INNER, D_MODEL, 0>
      <<<dim3(D_MODEL / 64, BL / 128), blk, 0, stream>>>(
          yg, W_out, out, nullptr);
}